// LocalGlobalRegistration_2173253452309
// MI455X (gfx1250) — compile-verified
//
#include <hip/hip_runtime.h>
#include <math.h>

#define BQ 1024
#define KQ 128
#define TOPKQ 3
#define MQ 8192
#define RADIUSQ 0.1f
#define THRESHQ 0.05f
#define MIN_LOCALQ 3
#define NUM_REFINEQ 5
#define EPSQ 1e-5f

#define NSORT (1 << 19)            // >= BQ*384 = 393216 candidate slots
#define CAND_PER_B (TOPKQ * KQ)    // 384 (multiple of 8)

#define O_GSRC 0
#define O_GTGT (MQ * 3)
#define O_GW   (MQ * 6)
#define O_T    (MQ * 7)

typedef float v2f __attribute__((ext_vector_type(2)));
typedef float v8f __attribute__((ext_vector_type(8)));

#if defined(__has_builtin)
#  if __has_builtin(__builtin_amdgcn_wmma_f32_16x16x4_f32)
#    define HAVE_WMMA_F32X4 1
#  endif
#endif

// ---------------------------------------------------------------- solvers ---

__device__ __forceinline__ float det3(const float M[3][3]) {
  return M[0][0] * (M[1][1] * M[2][2] - M[1][2] * M[2][1])
       - M[0][1] * (M[1][0] * M[2][2] - M[1][2] * M[2][0])
       + M[0][2] * (M[1][0] * M[2][1] - M[1][1] * M[2][0]);
}

// Weighted Kabsch: SVD(H)=U S V^T (via Jacobi on H^T H), R = V diag(1,1,d) U^T.
__device__ void kabsch_solve(float H[3][3], const float cs[3], const float ct[3],
                             float T[16]) {
  H[0][0] += 1e-6f; H[1][1] += 2e-6f; H[2][2] += 3e-6f;  // reference jitter
  float A[3][3];
  #pragma unroll
  for (int c = 0; c < 3; ++c)
    #pragma unroll
    for (int d = 0; d < 3; ++d)
      A[c][d] = H[0][c] * H[0][d] + H[1][c] * H[1][d] + H[2][c] * H[2][d];
  float V[3][3] = {{1.f, 0.f, 0.f}, {0.f, 1.f, 0.f}, {0.f, 0.f, 1.f}};
  const int PP[3] = {0, 0, 1}, QQ[3] = {1, 2, 2};
  for (int sweep = 0; sweep < 10; ++sweep) {
    #pragma unroll
    for (int pi = 0; pi < 3; ++pi) {
      int p = PP[pi], q = QQ[pi], r = 3 - p - q;
      float apq = A[p][q];
      if (fabsf(apq) < 1e-25f) continue;
      float theta = (A[q][q] - A[p][p]) / (2.f * apq);
      float tt = ((theta >= 0.f) ? 1.f : -1.f) /
                 (fabsf(theta) + sqrtf(theta * theta + 1.f));
      float cc = 1.f / sqrtf(tt * tt + 1.f);
      float ss = tt * cc;
      float app = A[p][p], aqq = A[q][q];
      A[p][p] = app - tt * apq; A[q][q] = aqq + tt * apq;
      A[p][q] = 0.f; A[q][p] = 0.f;
      float arp = A[r][p], arq = A[r][q];
      A[r][p] = cc * arp - ss * arq; A[p][r] = A[r][p];
      A[r][q] = ss * arp + cc * arq; A[q][r] = A[r][q];
      #pragma unroll
      for (int kk = 0; kk < 3; ++kk) {
        float vp = V[kk][p], vq = V[kk][q];
        V[kk][p] = cc * vp - ss * vq;
        V[kk][q] = ss * vp + cc * vq;
      }
    }
  }
  float lam[3] = {A[0][0], A[1][1], A[2][2]};
  #define SWAPEV(a, b)                                                         \
    if (lam[b] > lam[a]) {                                                     \
      float tl = lam[a]; lam[a] = lam[b]; lam[b] = tl;                         \
      for (int kk = 0; kk < 3; ++kk) {                                         \
        float tv = V[kk][a]; V[kk][a] = V[kk][b]; V[kk][b] = tv;               \
      }                                                                        \
    }
  SWAPEV(0, 1); SWAPEV(0, 2); SWAPEV(1, 2);
  #undef SWAPEV
  float U[3][3];
  #pragma unroll
  for (int k = 0; k < 3; ++k) {
    float ux = H[0][0] * V[0][k] + H[0][1] * V[1][k] + H[0][2] * V[2][k];
    float uy = H[1][0] * V[0][k] + H[1][1] * V[1][k] + H[1][2] * V[2][k];
    float uz = H[2][0] * V[0][k] + H[2][1] * V[1][k] + H[2][2] * V[2][k];
    float n = sqrtf(ux * ux + uy * uy + uz * uz);
    if (n > 1e-18f) {
      ux /= n; uy /= n; uz /= n;
    } else if (k == 2) {
      float cx = U[1][0] * U[2][1] - U[2][0] * U[1][1];
      float cy = U[2][0] * U[0][1] - U[0][0] * U[2][1];
      float cz = U[0][0] * U[1][1] - U[1][0] * U[0][1];
      float cn = sqrtf(cx * cx + cy * cy + cz * cz) + 1e-30f;
      ux = cx / cn; uy = cy / cn; uz = cz / cn;
    } else {
      ux = (k == 0) ? 1.f : 0.f; uy = (k == 1) ? 1.f : 0.f; uz = 0.f;
    }
    U[0][k] = ux; U[1][k] = uy; U[2][k] = uz;
  }
  float dd = det3(V) * det3(U);
  float R[3][3];
  #pragma unroll
  for (int a = 0; a < 3; ++a)
    #pragma unroll
    for (int b2 = 0; b2 < 3; ++b2)
      R[a][b2] = V[a][0] * U[b2][0] + V[a][1] * U[b2][1] + dd * V[a][2] * U[b2][2];
  #pragma unroll
  for (int a = 0; a < 3; ++a) {
    T[a * 4 + 0] = R[a][0]; T[a * 4 + 1] = R[a][1]; T[a * 4 + 2] = R[a][2];
    T[a * 4 + 3] = ct[a] - (R[a][0] * cs[0] + R[a][1] * cs[1] + R[a][2] * cs[2]);
  }
  T[12] = 0.f; T[13] = 0.f; T[14] = 0.f; T[15] = 1.f;
}

// Global weighted_procrustes from raw moments (exact centered form).
__device__ void solve_global_from_red(const float* r16, float* T) {
  float Wr = r16[0];
  float Wc = fmaxf(Wr, EPSQ);
  float csN[3] = {r16[1], r16[2], r16[3]};
  float ctN[3] = {r16[4], r16[5], r16[6]};
  float cs[3], ct[3];
  #pragma unroll
  for (int a = 0; a < 3; ++a) { cs[a] = csN[a] / Wc; ct[a] = ctN[a] / Wc; }
  float H[3][3];
  #pragma unroll
  for (int c = 0; c < 3; ++c)
    #pragma unroll
    for (int d = 0; d < 3; ++d)
      H[c][d] = r16[7 + c * 3 + d] - cs[c] * ctN[d] - csN[c] * ct[d]
                + Wr * cs[c] * ct[d];
  kabsch_solve(H, cs, ct, T);
}

// ---------------------------------------------- kernel A: score + moments ---
// Per batch: exp, row/col top-3, mutual-corr candidates compacted
// deterministically into transposed LDS staging:
//   candAT[c][n] = w_n * [src_i | 1][c]   (A side, w folded in)
//   candBT[d][n] = [tgt_j | 1][d]         (B side)
// zero-padded to a multiple of 8 rows. The 4x4 augmented moment
// M4 = P^T W Q is then a pure LDS-fed V_WMMA_F32_16X16X4_F32 reduction
// over candidates: per chunk one aligned ds_load_b64 per operand, no
// masking (lanes 4..15 read duplicate columns; they only feed D rows/cols
// >= 4, which are never extracted), no EXEC manipulation.

__global__ __launch_bounds__(128) void k_score(
    const float* __restrict__ src, const float* __restrict__ tgt,
    const unsigned char* __restrict__ smask, const unsigned char* __restrict__ tmask,
    const float* __restrict__ score,
    unsigned long long* __restrict__ keys, unsigned int* __restrict__ candCount,
    float* __restrict__ M4out, unsigned int* __restrict__ cntOut) {
  const int b = blockIdx.x;
  const int t = threadIdx.x;
  __shared__ int rowtop[KQ][3];
  __shared__ int coltop[KQ][3];
  __shared__ float candAT[4][CAND_PER_B];
  __shared__ float candBT[4][CAND_PER_B];
  __shared__ int counts[KQ];
  __shared__ int offs[KQ];
  __shared__ int ncs;
  __shared__ float partial[4][16];

  const float* srow = score + ((size_t)b * KQ + t) * KQ;
  // Row top-3 (strict > keeps lowest index on ties, matching lax.top_k).
  float v0 = -1e30f, v1 = -1e30f, v2 = -1e30f;
  int i0 = 0, i1 = 0, i2 = 0;
  for (int j = 0; j < KQ; ++j) {
    __builtin_prefetch(srow + j + 16, 0, 1);
    float s = expf(srow[j]);
    if (s > v0)      { v2 = v1; i2 = i1; v1 = v0; i1 = i0; v0 = s; i0 = j; }
    else if (s > v1) { v2 = v1; i2 = i1; v1 = s; i1 = j; }
    else if (s > v2) { v2 = s; i2 = j; }
  }
  rowtop[t][0] = i0; rowtop[t][1] = i1; rowtop[t][2] = i2;
  // Column top-3 (coalesced: row-contiguous reads across lanes).
  v0 = v1 = v2 = -1e30f; i0 = i1 = i2 = 0;
  const float* scol = score + (size_t)b * KQ * KQ + t;
  for (int i = 0; i < KQ; ++i) {
    float s = expf(scol[(size_t)i * KQ]);
    if (s > v0)      { v2 = v1; i2 = i1; v1 = v0; i1 = i0; v0 = s; i0 = i; }
    else if (s > v1) { v2 = v1; i2 = i1; v1 = s; i1 = i; }
    else if (s > v2) { v2 = s; i2 = i; }
  }
  coltop[t][0] = i0; coltop[t][1] = i1; coltop[t][2] = i2;
  __syncthreads();

  // Mutual-top3 candidates of row t.
  int myj[3]; float mys[3]; int myc = 0;
  bool sm = smask[b * KQ + t] != 0;
  #pragma unroll
  for (int r = 0; r < 3; ++r) {
    int j = rowtop[t][r];
    float s = expf(srow[j]);
    bool pass = sm && (tmask[b * KQ + j] != 0) && (s > THRESHQ) &&
                (coltop[j][0] == t || coltop[j][1] == t || coltop[j][2] == t);
    if (pass) { myj[myc] = j; mys[myc] = s; ++myc; }
  }
  counts[t] = myc;
  __syncthreads();
  if (t == 0) {
    int acc = 0;
    for (int i = 0; i < KQ; ++i) { offs[i] = acc; acc += counts[i]; }
    ncs = acc;
  }
  __syncthreads();
  // Stage operands (src row reused across this thread's candidates).
  float sx = src[((size_t)b * KQ + t) * 3 + 0];
  float sy = src[((size_t)b * KQ + t) * 3 + 1];
  float sz = src[((size_t)b * KQ + t) * 3 + 2];
  for (int c = 0; c < myc; ++c) {
    int idx = offs[t] + c;
    float w = mys[c];
    int j = myj[c];
    candAT[0][idx] = w * sx; candAT[1][idx] = w * sy;
    candAT[2][idx] = w * sz; candAT[3][idx] = w;
    candBT[0][idx] = tgt[((size_t)b * KQ + j) * 3 + 0];
    candBT[1][idx] = tgt[((size_t)b * KQ + j) * 3 + 1];
    candBT[2][idx] = tgt[((size_t)b * KQ + j) * 3 + 2];
    candBT[3][idx] = 1.f;
    unsigned flat = ((unsigned)b << 14) | ((unsigned)t << 7) | (unsigned)j;
    unsigned long long key =
        ((unsigned long long)__float_as_uint(w) << 24) |
        (unsigned long long)((~flat) & 0xFFFFFFu);
    keys[(size_t)b * CAND_PER_B + idx] = key;  // fixed slot: deterministic
  }
  if (t == 0) {
    cntOut[b] = (unsigned)ncs;
    atomicAdd(candCount, (unsigned)ncs);
  }
  __syncthreads();
  const int nc = ncs;
  const int padN = (nc + 7) & ~7;  // <= CAND_PER_B (384 is a multiple of 8)
  for (int r = nc + t; r < padN; r += 128) {
    candAT[0][r] = 0.f; candAT[1][r] = 0.f; candAT[2][r] = 0.f; candAT[3][r] = 0.f;
    candBT[0][r] = 0.f; candBT[1][r] = 0.f; candBT[2][r] = 0.f; candBT[3][r] = 0.f;
  }
  __syncthreads();

#ifdef HAVE_WMMA_F32X4
  // A tile 16x4 f32: lane = 16*(k>=2)+row, vgpr = k&1. B tile 4x16: lane = 16*(k>=2)+col.
  // Contiguous K-pairs -> one ds_load_b64 per operand per chunk.
  const int wave = t >> 5;
  const int lane = t & 31;
  const int half = lane >> 4;
  const int ln = lane & 15;
  const int cidx = ln & 3;  // lanes 4..15 duplicate cols: only feed D rows/cols >=4
  v8f acc = {0.f, 0.f, 0.f, 0.f, 0.f, 0.f, 0.f, 0.f};
  const int chunks = padN >> 2;  // even
  for (int ch = wave * 2; ch < chunks; ch += 8) {
    int r0 = ch * 4 + 2 * half;  // even -> 8B aligned in LDS
    v2f av = *(const v2f*)&candAT[cidx][r0];
    v2f bv = *(const v2f*)&candBT[cidx][r0];
    acc = __builtin_amdgcn_wmma_f32_16x16x4_f32(false, av, false, bv,
                                                (short)0, acc, false, false);
    int r1 = r0 + 4;
    v2f aw = *(const v2f*)&candAT[cidx][r1];
    v2f bw = *(const v2f*)&candBT[cidx][r1];
    acc = __builtin_amdgcn_wmma_f32_16x16x4_f32(false, aw, false, bw,
                                                (short)0, acc, false, false);
  }
  // C/D 16x16 f32: vgpr r, lanes 0-15 -> row r, col=lane. M4 rows 0..3 live there.
  if (half == 0 && ln < 4) {
    #pragma unroll
    for (int c = 0; c < 4; ++c) partial[wave][c * 4 + ln] = acc[c];
  }
  __syncthreads();
  if (t < 16)
    M4out[(size_t)b * 16 + t] =
        partial[0][t] + partial[1][t] + partial[2][t] + partial[3][t];
#else
  if (t == 0) {
    float mloc[16];
    #pragma unroll
    for (int k = 0; k < 16; ++k) mloc[k] = 0.f;
    for (int n = 0; n < nc; ++n)
      for (int c = 0; c < 4; ++c)
        for (int d = 0; d < 4; ++d) mloc[c * 4 + d] += candAT[c][n] * candBT[d][n];
    for (int k = 0; k < 16; ++k) M4out[(size_t)b * 16 + k] = mloc[k];
  }
#endif
}

// ------------------------------------------------------- bitonic sort step ---

__global__ __launch_bounds__(256) void k_bitonic(unsigned long long* __restrict__ a,
                                                 int j, int k) {
  unsigned i = blockIdx.x * 256u + threadIdx.x;
  unsigned ixj = i ^ (unsigned)j;
  if (ixj > i) {
    unsigned long long x = a[i], y = a[ixj];
    bool up = ((i & (unsigned)k) == 0);  // descending overall
    bool sw = up ? (x < y) : (x > y);
    if (sw) { a[i] = y; a[ixj] = x; }
  }
}

// -------------------------------------------- top-M scatter + zero-pad fill ---

__global__ __launch_bounds__(256) void k_scatter(const unsigned long long* __restrict__ keys,
                                                 const float* __restrict__ src,
                                                 const float* __restrict__ tgt,
                                                 float* __restrict__ out) {
  int m = blockIdx.x * 256 + threadIdx.x;
  if (m >= MQ) return;
  unsigned long long key = keys[m];
  if (key == 0ull) return;  // padded slot, handled by k_fill
  unsigned flat = (~(unsigned)(key & 0xFFFFFFull)) & 0xFFFFFFu;
  float s = __uint_as_float((unsigned)(key >> 24));
  int b = flat >> 14, i = (flat >> 7) & 127, j = flat & 127;
  out[O_GW + m] = s;
  #pragma unroll
  for (int c = 0; c < 3; ++c) {
    out[O_GSRC + m * 3 + c] = src[((size_t)b * KQ + i) * 3 + c];
    out[O_GTGT + m * 3 + c] = tgt[((size_t)b * KQ + j) * 3 + c];
  }
}

// Padded top_k slots select the earliest flat zero positions (all in batch 0).
__global__ __launch_bounds__(128) void k_fill(
    const float* __restrict__ src, const float* __restrict__ tgt,
    const unsigned char* __restrict__ smask, const unsigned char* __restrict__ tmask,
    const float* __restrict__ score, const unsigned int* __restrict__ candCount,
    float* __restrict__ out) {
  __shared__ int rowtop[KQ][3];
  __shared__ int coltop[KQ][3];
  const int t = threadIdx.x;
  const float* srow = score + (size_t)t * KQ;
  float v0 = -1e30f, v1 = -1e30f, v2 = -1e30f;
  int i0 = 0, i1 = 0, i2 = 0;
  for (int j = 0; j < KQ; ++j) {
    float s = expf(srow[j]);
    if (s > v0)      { v2 = v1; i2 = i1; v1 = v0; i1 = i0; v0 = s; i0 = j; }
    else if (s > v1) { v2 = v1; i2 = i1; v1 = s; i1 = j; }
    else if (s > v2) { v2 = s; i2 = j; }
  }
  rowtop[t][0] = i0; rowtop[t][1] = i1; rowtop[t][2] = i2;
  v0 = v1 = v2 = -1e30f; i0 = i1 = i2 = 0;
  for (int i = 0; i < KQ; ++i) {
    float s = expf(score[(size_t)i * KQ + t]);
    if (s > v0)      { v2 = v1; i2 = i1; v1 = v0; i1 = i0; v0 = s; i0 = i; }
    else if (s > v1) { v2 = v1; i2 = i1; v1 = s; i1 = i; }
    else if (s > v2) { v2 = s; i2 = i; }
  }
  coltop[t][0] = i0; coltop[t][1] = i1; coltop[t][2] = i2;
  __syncthreads();
  if (t != 0) return;
  unsigned cc = *candCount;
  int nnz = (cc < (unsigned)MQ) ? (int)cc : MQ;
  int need = MQ - nnz;
  if (need <= 0) return;
  int r = 0;
  for (int f = 0; f < KQ * KQ && r < need; ++f) {
    int i = f >> 7, j = f & 127;
    float s = expf(score[f]);
    bool isc = (smask[i] != 0) && (tmask[j] != 0) && (s > THRESHQ) &&
               (rowtop[i][0] == j || rowtop[i][1] == j || rowtop[i][2] == j) &&
               (coltop[j][0] == i || coltop[j][1] == i || coltop[j][2] == i);
    if (!isc) {
      int slot = nnz + r;
      out[O_GW + slot] = 0.f;
      #pragma unroll
      for (int c = 0; c < 3; ++c) {
        out[O_GSRC + slot * 3 + c] = src[i * 3 + c];
        out[O_GTGT + slot * 3 + c] = tgt[j * 3 + c];
      }
      ++r;
    }
  }
}

// ---------------------------------------------- kernel C: per-patch solves ---

__global__ __launch_bounds__(128) void k_patch(const float* __restrict__ M4,
                                               float* __restrict__ Tp) {
  int b = blockIdx.x * 128 + threadIdx.x;
  if (b >= BQ) return;
  const float* m = M4 + (size_t)b * 16;
  float Wc = fmaxf(m[15], EPSQ);
  float cs[3], ct[3];
  #pragma unroll
  for (int c = 0; c < 3; ++c) { cs[c] = m[c * 4 + 3] / Wc; ct[c] = m[12 + c] / Wc; }
  float H[3][3];
  #pragma unroll
  for (int c = 0; c < 3; ++c)
    #pragma unroll
    for (int d = 0; d < 3; ++d)
      H[c][d] = m[c * 4 + d] - Wc * cs[c] * ct[d];  // reference patch form
  float T[16];
  kabsch_solve(H, cs, ct, T);
  #pragma unroll
  for (int k = 0; k < 16; ++k) Tp[(size_t)b * 16 + k] = T[k];
}

// ----------------------------------------------- kernel D: inlier counting ---

__global__ __launch_bounds__(256) void k_inl(const float* __restrict__ Tp,
                                             const unsigned int* __restrict__ cnt,
                                             const float* __restrict__ out,
                                             int* __restrict__ inl) {
  const int b = blockIdx.x, t = threadIdx.x;
  __shared__ float Ts[16];
  __shared__ int red[256];
  if (t < 16) Ts[t] = Tp[(size_t)b * 16 + t];
  __syncthreads();
  int loc = 0;
  for (int m = t; m < MQ; m += 256) {
    float w = out[O_GW + m];
    if (w > 0.f) {
      float sx = out[O_GSRC + m * 3], sy = out[O_GSRC + m * 3 + 1], sz = out[O_GSRC + m * 3 + 2];
      float tx = out[O_GTGT + m * 3], ty = out[O_GTGT + m * 3 + 1], tz = out[O_GTGT + m * 3 + 2];
      float ax = Ts[0] * sx + Ts[1] * sy + Ts[2] * sz + Ts[3];
      float ay = Ts[4] * sx + Ts[5] * sy + Ts[6] * sz + Ts[7];
      float az = Ts[8] * sx + Ts[9] * sy + Ts[10] * sz + Ts[11];
      float dx = tx - ax, dy = ty - ay, dz = tz - az;
      if (sqrtf(dx * dx + dy * dy + dz * dz + 1e-12f) < RADIUSQ) ++loc;
    }
  }
  red[t] = loc;
  __syncthreads();
  for (int s = 128; s > 0; s >>= 1) {
    if (t < s) red[t] += red[t + s];
    __syncthreads();
  }
  if (t == 0) inl[b] = (cnt[b] >= MIN_LOCALQ) ? red[0] : -1;
}

// --------------------------------------- kernel E: argmax + global + refine ---

__device__ void accum_moments(const float* __restrict__ out, const float* Tcur,
                              float* red, int t) {
  __syncthreads();
  float loc[16];
  #pragma unroll
  for (int k = 0; k < 16; ++k) loc[k] = 0.f;
  for (int m = t; m < MQ; m += 256) {
    float w = out[O_GW + m];
    if (w > 0.f) {
      float sx = out[O_GSRC + m * 3], sy = out[O_GSRC + m * 3 + 1], sz = out[O_GSRC + m * 3 + 2];
      float tx = out[O_GTGT + m * 3], ty = out[O_GTGT + m * 3 + 1], tz = out[O_GTGT + m * 3 + 2];
      if (Tcur) {
        float ax = Tcur[0] * sx + Tcur[1] * sy + Tcur[2] * sz + Tcur[3];
        float ay = Tcur[4] * sx + Tcur[5] * sy + Tcur[6] * sz + Tcur[7];
        float az = Tcur[8] * sx + Tcur[9] * sy + Tcur[10] * sz + Tcur[11];
        float dx = tx - ax, dy = ty - ay, dz = tz - az;
        if (!(sqrtf(dx * dx + dy * dy + dz * dz + 1e-12f) < RADIUSQ)) w = 0.f;
      }
      if (w > 0.f) {
        loc[0] += w;
        loc[1] += w * sx; loc[2] += w * sy; loc[3] += w * sz;
        loc[4] += w * tx; loc[5] += w * ty; loc[6] += w * tz;
        loc[7]  += w * sx * tx; loc[8]  += w * sx * ty; loc[9]  += w * sx * tz;
        loc[10] += w * sy * tx; loc[11] += w * sy * ty; loc[12] += w * sy * tz;
        loc[13] += w * sz * tx; loc[14] += w * sz * ty; loc[15] += w * sz * tz;
      }
    }
  }
  #pragma unroll
  for (int k = 0; k < 16; ++k) red[t * 16 + k] = loc[k];
  __syncthreads();
  for (int s = 128; s > 0; s >>= 1) {
    if (t < s) {
      #pragma unroll
      for (int k = 0; k < 16; ++k) red[t * 16 + k] += red[(t + s) * 16 + k];
    }
    __syncthreads();
  }
}

__global__ __launch_bounds__(256) void k_final(const float* __restrict__ Tp,
                                               const int* __restrict__ inl,
                                               float* __restrict__ out) {
  __shared__ float red[256 * 16];
  __shared__ float Ts[16];
  __shared__ float Tall[16];
  __shared__ int bv[256], bidx[256];
  __shared__ int bestIdx, anyValid;
  const int t = threadIdx.x;

  int lv = -2, li = 0;
  for (int b = t; b < BQ; b += 256) {
    int v = inl[b];
    if (v > lv) { lv = v; li = b; }  // strict >: first max wins per thread
  }
  bv[t] = lv; bidx[t] = li;
  __syncthreads();
  for (int s = 128; s > 0; s >>= 1) {
    if (t < s) {
      if (bv[t + s] > bv[t] || (bv[t + s] == bv[t] && bidx[t + s] < bidx[t])) {
        bv[t] = bv[t + s]; bidx[t] = bidx[t + s];
      }
    }
    __syncthreads();
  }
  if (t == 0) { bestIdx = bidx[0]; anyValid = (bv[0] >= 0); }
  __syncthreads();

  accum_moments(out, nullptr, red, t);  // T_all fallback fit (w = g_w)
  if (t == 0) solve_global_from_red(red, Tall);
  __syncthreads();
  if (t == 0) {
    for (int k = 0; k < 16; ++k)
      Ts[k] = anyValid ? Tp[(size_t)bestIdx * 16 + k] : Tall[k];
  }
  __syncthreads();
  for (int it = 0; it < NUM_REFINEQ; ++it) {
    accum_moments(out, Ts, red, t);
    if (t == 0) solve_global_from_red(red, Ts);
    __syncthreads();
  }
  if (t < 16) out[O_T + t] = Ts[t];
}

// ------------------------------------------------------------------ launch ---

extern "C" void kernel_launch(void* const* d_in, const int* in_sizes, int n_in,
                              void* d_out, int out_size, void* d_ws, size_t ws_size,
                              hipStream_t stream) {
  (void)in_sizes; (void)n_in; (void)out_size; (void)ws_size;
  const float* src = (const float*)d_in[0];
  const float* tgt = (const float*)d_in[1];
  const unsigned char* sm = (const unsigned char*)d_in[2];
  const unsigned char* tm = (const unsigned char*)d_in[3];
  const float* score = (const float*)d_in[4];
  float* out = (float*)d_out;

  char* ws = (char*)d_ws;
  unsigned int* candCount = (unsigned int*)ws;                       // 16 B
  unsigned long long* keys = (unsigned long long*)(ws + 16);         // 4 MB
  float* M4 = (float*)(ws + 16 + (size_t)NSORT * 8);                 // B*16 f32
  unsigned int* cnt = (unsigned int*)((char*)M4 + (size_t)BQ * 16 * 4);
  float* Tp = (float*)((char*)cnt + (size_t)BQ * 4);
  int* inl = (int*)((char*)Tp + (size_t)BQ * 16 * 4);

  // Re-init counter + key slots every call (deterministic replays).
  hipMemsetAsync(d_ws, 0, 16 + (size_t)NSORT * 8, stream);

  k_score<<<BQ, 128, 0, stream>>>(src, tgt, sm, tm, score, keys, candCount, M4, cnt);

  for (int k = 2; k <= NSORT; k <<= 1)
    for (int j = k >> 1; j > 0; j >>= 1)
      k_bitonic<<<NSORT / 256, 256, 0, stream>>>(keys, j, k);

  k_scatter<<<MQ / 256, 256, 0, stream>>>(keys, src, tgt, out);
  k_fill<<<1, 128, 0, stream>>>(src, tgt, sm, tm, score, candCount, out);
  k_patch<<<(BQ + 127) / 128, 128, 0, stream>>>(M4, Tp);
  k_inl<<<BQ, 256, 0, stream>>>(Tp, cnt, out, inl);
  k_final<<<1, 256, 0, stream>>>(Tp, inl, out);
}